// Cvx_Nets_3487513444587
// MI455X (gfx1250) — compile-verified
//
#include <hip/hip_runtime.h>

typedef float v2f __attribute__((ext_vector_type(2)));
typedef float v8f __attribute__((ext_vector_type(8)));

namespace {
constexpr int kNZ = 10;     // u variables
constexpr int kNW = 90;     // total primal vars
constexpr int kMI = 180;    // inequality rows
constexpr int kN  = 91;     // KKT dimension (NW + 1 equality)
constexpr int kKS = 97;     // LDS row stride for KKT (odd -> conflict-friendly)
constexpr int kGR = 192;    // padded G rows (K dimension, 180 -> 192)
constexpr int kGC = 96;     // padded G cols (90 -> 96)
constexpr int kHD = 96;     // padded H dimension
constexpr int kIters = 20;
constexpr int kThreads = 192;   // 6 wave32 per block
constexpr float kSigma = 0.1f;
constexpr float kReg = 1e-8f;

// workspace layout (float offsets)
constexpr int OFF_G  = 0;                    // padded G: 192 x 96
constexpr int OFF_H  = OFF_G + kGR * kGC;    // padded H + REG*I: 96 x 96
constexpr int OFF_QH = OFF_H + kHD * kHD;    // Q_hat: 10 x 10
constexpr int OFF_Q  = OFF_QH + 100;         // Q: 4 x 4
constexpr int OFF_AH = OFF_Q + 16;           // A_hat: 40 x 4
constexpr int OFF_QB = OFF_AH + 160;         // Q_diag @ B_hat: 40 x 10
} // namespace

// ---------------------------------------------------------------------------
// Setup kernel: build batch-independent matrices into ws (single block).
// ---------------------------------------------------------------------------
__global__ __launch_bounds__(256) void mpc_setup_kernel(
    const float* __restrict__ Qs, const float* __restrict__ Rs,
    const float* __restrict__ Ad, const float* __restrict__ Bd,
    float* __restrict__ ws)
{
  __shared__ float Apow[10 * 16];   // A^1..A^10
  __shared__ float T[10 * 4];       // A^j B
  __shared__ float Bh[40 * 10];     // B_hat
  __shared__ float Qsh[16];
  __shared__ float QBsh[400];
  __shared__ float Qh[100];
  const int tid = threadIdx.x;

  if (tid == 0) {
    for (int e = 0; e < 16; ++e) Apow[e] = Ad[e];
    for (int p = 1; p < 10; ++p)
      for (int r = 0; r < 4; ++r)
        for (int c = 0; c < 4; ++c) {
          float acc = 0.f;
          for (int k = 0; k < 4; ++k) acc += Apow[(p - 1) * 16 + r * 4 + k] * Ad[k * 4 + c];
          Apow[p * 16 + r * 4 + c] = acc;
        }
    for (int r = 0; r < 4; ++r) T[r] = Bd[r];
    for (int j = 1; j < 10; ++j)
      for (int r = 0; r < 4; ++r) {
        float acc = 0.f;
        for (int k = 0; k < 4; ++k) acc += Ad[r * 4 + k] * T[(j - 1) * 4 + k];
        T[j * 4 + r] = acc;
      }
  }
  __syncthreads();

  for (int e = tid; e < 16; e += 256) {           // Q = Qs Qs^T
    int r = e >> 2, c = e & 3;
    float acc = 0.f;
    for (int k = 0; k < 4; ++k) acc += Qs[r * 4 + k] * Qs[c * 4 + k];
    Qsh[e] = acc; ws[OFF_Q + e] = acc;
  }
  for (int e = tid; e < 400; e += 256) {          // B_hat (40 x 10)
    int row = e / 10, j = e % 10;
    int i = row >> 2, r = row & 3;
    Bh[e] = (i >= j) ? T[(i - j) * 4 + r] : 0.f;
  }
  for (int e = tid; e < 160; e += 256) {          // A_hat (40 x 4)
    int row = e >> 2, c = e & 3;
    ws[OFF_AH + e] = Apow[(row >> 2) * 16 + (row & 3) * 4 + c];
  }
  __syncthreads();

  for (int e = tid; e < 400; e += 256) {          // QB = Q_diag @ B_hat
    int row = e / 10, j = e % 10;
    int i = row >> 2, r = row & 3;
    float acc = 0.f;
    for (int c = 0; c < 4; ++c) acc += Qsh[r * 4 + c] * Bh[(i * 4 + c) * 10 + j];
    QBsh[e] = acc; ws[OFF_QB + e] = acc;
  }
  __syncthreads();

  const float Rv = Rs[0] * Rs[0];
  for (int e = tid; e < 100; e += 256) {          // Q_hat = B_hat^T QB + R I
    int a = e / 10, b = e % 10;
    float acc = (a == b) ? Rv : 0.f;
    for (int k = 0; k < 40; ++k) acc += Bh[k * 10 + a] * QBsh[k * 10 + b];
    Qh[e] = acc; ws[OFF_QH + e] = acc;
  }
  __syncthreads();

  for (int e = tid; e < kHD * kHD; e += 256) {    // Hreg = H + REG*I, padded
    int r = e / kHD, c = e % kHD;
    float v = 0.f;
    if (r < 10 && c < 10) v = 2.f * Qh[r * 10 + c];
    else if (r == c && r < kNW) v = 2.f;
    if (r == c && r < kNW) v += kReg;
    ws[OFF_H + e] = v;
  }
  for (int e = tid; e < kGR * kGC; e += 256) {    // padded G_all (192 x 96)
    int k = e / kGC, j = e % kGC;
    float v = 0.f;
    if (k < 10) { if (j == k) v = 1.f; }
    else if (k < 20) { if (j == k - 10) v = -1.f; }
    else if (k < 100) {
      int r = k - 20;
      if (j < 10) v = (r < 40) ? Bh[r * 10 + j] : -Bh[(r - 40) * 10 + j];
      if (j == 10 + r) v = -1.f;
    } else if (k < 180) {
      int r = k - 100;
      if (j == 10 + r) v = -1.f;
    }
    ws[OFF_G + e] = v;
  }
}

// ---------------------------------------------------------------------------
// IPM solver: one workgroup (6 wave32) per batch element.
// Wave w owns tile-row w of the 96x96 M matrix: 6 independent WMMA
// accumulator chains per wave, A fragment reused across all 6 column tiles.
// ---------------------------------------------------------------------------
__global__ __launch_bounds__(kThreads) void mpc_ipm_kernel(
    const float* __restrict__ x_all, const float* __restrict__ u0_all,
    const float* __restrict__ ws, float* __restrict__ out, int batch)
{
  const int b = blockIdx.x;
  const int tid = threadIdx.x;
  const int lane = tid & 31;
  const int wave = tid >> 5;    // 0..5 = tile row
  const int half = lane >> 4;   // 0: lanes 0-15, 1: lanes 16-31
  const int l16 = lane & 15;

  const float* __restrict__ gG  = ws + OFF_G;
  const float* __restrict__ gH  = ws + OFF_H;
  const float* __restrict__ gQH = ws + OFF_QH;
  const float* __restrict__ gQ  = ws + OFF_Q;
  const float* __restrict__ gAH = ws + OFF_AH;
  const float* __restrict__ gQB = ws + OFF_QB;

  __shared__ float Kls[96 * kKS];        // KKT + rhs (col 91)
  __shared__ float w_s[kNW];
  __shared__ float s_s[kMI], lam_s[kMI], d_s[kGR], tmp_s[kMI];
  __shared__ float rprim_s[kMI], dsv_s[kMI], dlam_s[kMI];
  __shared__ float rw_s[kNW], sol_s[kN];
  __shared__ float h_s[kMI], p_s[kNZ], Qh_s[100], Ax0_s[40], Q_s[16], x_s[4];
  __shared__ float red_s[kThreads];
  __shared__ float sc_nu, sc_beq, sc_rn;

  // -------- prologue: per-batch data --------
  if (tid < 4) x_s[tid] = x_all[b * 4 + tid];
  if (tid == 0) { sc_beq = u0_all[b]; sc_nu = 0.f; }
  for (int e = tid; e < 100; e += kThreads) Qh_s[e] = gQH[e];
  for (int e = tid; e < 16; e += kThreads) Q_s[e] = gQ[e];
  __syncthreads();

  for (int i = tid; i < 40; i += kThreads) {     // A_x0 = A_hat @ x
    float acc = 0.f;
    for (int c = 0; c < 4; ++c) acc += gAH[i * 4 + c] * x_s[c];
    Ax0_s[i] = acc;
  }
  __syncthreads();

  for (int j = tid; j < kNZ; j += kThreads) {    // p = 2 A_x0 @ QB
    float acc = 0.f;
    for (int i = 0; i < 40; ++i) acc += Ax0_s[i] * gQB[i * 10 + j];
    p_s[j] = 2.f * acc;
  }
  for (int k = tid; k < kMI; k += kThreads) {    // h_all, s, lam init
    float h;
    if (k < 20) h = 0.5f;
    else if (k < 60) h = 4.f - Ax0_s[k - 20];
    else if (k < 100) h = 4.f + Ax0_s[k - 60];
    else h = 0.f;
    h_s[k] = h;
    s_s[k] = fmaxf(h, 1.f);
    lam_s[k] = 1.f;
  }
  for (int j = tid; j < kNW; j += kThreads) w_s[j] = 0.f;
  for (int k = kMI + tid; k < kGR; k += kThreads) d_s[k] = 0.f;  // K padding
  __syncthreads();

  for (int it = 0; it < kIters; ++it) {
    // -------- residuals, mu, d, tmp --------
    float part = 0.f;
    for (int k = tid; k < kMI; k += kThreads) {
      const float* grow = gG + k * kGC;
      float acc = 0.f;
      for (int j = 0; j < kNW; ++j) acc += grow[j] * w_s[j];
      rprim_s[k] = acc + s_s[k] - h_s[k];
      part += s_s[k] * lam_s[k];
    }
    red_s[tid] = part;
    __syncthreads();
    if (tid < 64) red_s[tid] += red_s[tid + 128];   // 192 -> 128
    __syncthreads();
    for (int off = 64; off > 0; off >>= 1) {
      if (tid < off) red_s[tid] += red_s[tid + off];
      __syncthreads();
    }
    const float mu = kSigma * red_s[0] / (float)kMI;

    for (int k = tid; k < kMI; k += kThreads) {
      const float sk = s_s[k], lk = lam_s[k];
      d_s[k] = lk / sk;
      tmp_s[k] = (mu - lk * sk + lk * rprim_s[k]) / sk;
    }
    __syncthreads();

    for (int j = tid; j < kNW; j += kThreads) {  // rw = -(r_dual + tmp @ G)
      float accL = 0.f, accT = 0.f;
      for (int k = 0; k < kMI; ++k) {
        const float g = gG[k * kGC + j];
        accL += lam_s[k] * g;
        accT += tmp_s[k] * g;
      }
      float hw;
      if (j < kNZ) {
        float a2 = 0.f;
        for (int i = 0; i < kNZ; ++i) a2 += w_s[i] * Qh_s[i * 10 + j];
        hw = 2.f * a2;
      } else hw = 2.f * w_s[j];
      const float qj = (j < kNZ) ? p_s[j] : 0.f;
      const float rd = hw + qj + accL + ((j == 0) ? sc_nu : 0.f);
      rw_s[j] = -(rd + accT);
    }
    if (tid == 0) sc_rn = -(w_s[0] - sc_beq);
    __syncthreads();

    // -------- M = Hreg + G^T diag(d) G via fp32 WMMA --------
    // Wave `wave` computes tile row m0=wave*16 against all 6 column tiles:
    // one shared A fragment per K-step, 6 independent accumulators.
    {
      const int m0 = wave * 16;
      const int mrow = m0 + half * 8;
      v8f c0, c1, c2, c3, c4, c5;
#pragma unroll
      for (int r = 0; r < 8; ++r) {
        const float* hrow = gH + (mrow + r) * kHD + l16;
        c0[r] = hrow[0];  c1[r] = hrow[16]; c2[r] = hrow[32];
        c3[r] = hrow[48]; c4[r] = hrow[64]; c5[r] = hrow[80];
      }
      for (int kk = 0; kk < 48; ++kk) {
        const int k0 = kk * 4 + half * 2;
        const float* g0 = gG + k0 * kGC + l16;
        const float* g1 = g0 + kGC;
        const float dv0 = d_s[k0], dv1 = d_s[k0 + 1];
        v2f a;
        a[0] = g0[m0]; a[1] = g1[m0];
        v2f bb;
        bb[0] = g0[0]  * dv0; bb[1] = g1[0]  * dv1;
        c0 = __builtin_amdgcn_wmma_f32_16x16x4_f32(false, a, false, bb, (short)0, c0, false, false);
        bb[0] = g0[16] * dv0; bb[1] = g1[16] * dv1;
        c1 = __builtin_amdgcn_wmma_f32_16x16x4_f32(false, a, false, bb, (short)0, c1, false, false);
        bb[0] = g0[32] * dv0; bb[1] = g1[32] * dv1;
        c2 = __builtin_amdgcn_wmma_f32_16x16x4_f32(false, a, false, bb, (short)0, c2, false, false);
        bb[0] = g0[48] * dv0; bb[1] = g1[48] * dv1;
        c3 = __builtin_amdgcn_wmma_f32_16x16x4_f32(false, a, false, bb, (short)0, c3, false, false);
        bb[0] = g0[64] * dv0; bb[1] = g1[64] * dv1;
        c4 = __builtin_amdgcn_wmma_f32_16x16x4_f32(false, a, false, bb, (short)0, c4, false, false);
        bb[0] = g0[80] * dv0; bb[1] = g1[80] * dv1;
        c5 = __builtin_amdgcn_wmma_f32_16x16x4_f32(false, a, false, bb, (short)0, c5, false, false);
      }
#pragma unroll
      for (int r = 0; r < 8; ++r) {
        float* krow = Kls + (mrow + r) * kKS + l16;
        krow[0]  = c0[r]; krow[16] = c1[r]; krow[32] = c2[r];
        krow[48] = c3[r]; krow[64] = c4[r]; krow[80] = c5[r];
      }
    }
    __syncthreads();

    // -------- KKT borders (Aeq = e0) + rhs column --------
    for (int j = tid; j < kNW; j += kThreads) {
      Kls[j * kKS + kNW] = (j == 0) ? 1.f : 0.f;
      Kls[kNW * kKS + j] = (j == 0) ? 1.f : 0.f;
      Kls[j * kKS + kN]  = rw_s[j];
    }
    if (tid == 0) {
      Kls[kNW * kKS + kNW] = -kReg;
      Kls[kNW * kKS + kN]  = sc_rn;
    }
    __syncthreads();

    // -------- Gauss-Jordan elimination (no pivoting, no back-subst) ------
    // 2 threads per row (91 rows * 2 = 182 <= 192); one barrier per column.
    {
      const int i = tid >> 1;        // row
      const int hh = tid & 1;        // column parity
      for (int k = 0; k < kN; ++k) {
        if (i < kN && i != k) {
          const float piv = Kls[k * kKS + k];
          const float f = Kls[i * kKS + k] / piv;
          for (int j = k + 1 + hh; j <= kN; j += 2)   // includes rhs col
            Kls[i * kKS + j] -= f * Kls[k * kKS + j];
        }
        __syncthreads();
      }
    }
    if (tid < kN) sol_s[tid] = Kls[tid * kKS + kN] / Kls[tid * kKS + tid];
    __syncthreads();

    // -------- ds, dlam, fraction-to-boundary, update --------
    float pmin = 1.f;
    for (int k = tid; k < kMI; k += kThreads) {
      const float* grow = gG + k * kGC;
      float acc = 0.f;
      for (int j = 0; j < kNW; ++j) acc += grow[j] * sol_s[j];
      const float dsk = -acc - rprim_s[k];
      const float lk = lam_s[k], sk = s_s[k];
      const float dlk = (mu - lk * sk - lk * dsk) / sk;
      dsv_s[k] = dsk; dlam_s[k] = dlk;
      if (dsk < -1e-12f) pmin = fminf(pmin, -sk / fminf(dsk, -1e-12f));
      if (dlk < -1e-12f) pmin = fminf(pmin, -lk / fminf(dlk, -1e-12f));
    }
    red_s[tid] = pmin;
    __syncthreads();
    if (tid < 64) red_s[tid] = fminf(red_s[tid], red_s[tid + 128]);
    __syncthreads();
    for (int off = 64; off > 0; off >>= 1) {
      if (tid < off) red_s[tid] = fminf(red_s[tid], red_s[tid + off]);
      __syncthreads();
    }
    const float alpha = 0.99f * red_s[0];

    for (int j = tid; j < kNW; j += kThreads) w_s[j] += alpha * sol_s[j];
    for (int k = tid; k < kMI; k += kThreads) {
      s_s[k]   += alpha * dsv_s[k];
      lam_s[k] += alpha * dlam_s[k];
    }
    if (tid == 0) sc_nu += alpha * sol_s[kNW];
    __syncthreads();
  }

  // -------- outputs: cost_opt then u0_opt --------
  if (tid == 0) {
    float a = 0.f;
    for (int i = 0; i < kNZ; ++i) {
      float tq = 0.f;
      for (int j = 0; j < kNZ; ++j) tq += w_s[j] * Qh_s[j * 10 + i];
      a += (tq + p_s[i]) * w_s[i];
    }
    float bt = 0.f;
    for (int blk = 0; blk < 10; ++blk)
      for (int r = 0; r < 4; ++r)
        for (int c = 0; c < 4; ++c)
          bt += Ax0_s[blk * 4 + r] * Q_s[r * 4 + c] * Ax0_s[blk * 4 + c];
    float cc = 0.f;
    for (int r = 0; r < 4; ++r)
      for (int c = 0; c < 4; ++c)
        cc += x_s[r] * Q_s[r * 4 + c] * x_s[c];
    float dt = 0.f;
    for (int j = kNZ; j < kNW; ++j) dt += w_s[j] * w_s[j];
    out[b] = a + bt + cc + dt;
    out[batch + b] = w_s[0];
  }
}

extern "C" void kernel_launch(void* const* d_in, const int* in_sizes, int n_in,
                              void* d_out, int out_size, void* d_ws, size_t ws_size,
                              hipStream_t stream)
{
  const float* x  = (const float*)d_in[0];
  const float* u0 = (const float*)d_in[1];
  const float* Qs = (const float*)d_in[2];
  const float* Rs = (const float*)d_in[3];
  const float* Ad = (const float*)d_in[4];
  const float* Bd = (const float*)d_in[5];
  float* ws  = (float*)d_ws;
  float* out = (float*)d_out;
  const int batch = in_sizes[0] / 4;

  hipLaunchKernelGGL(mpc_setup_kernel, dim3(1), dim3(256), 0, stream,
                     Qs, Rs, Ad, Bd, ws);
  hipLaunchKernelGGL(mpc_ipm_kernel, dim3(batch), dim3(kThreads), 0, stream,
                     x, u0, ws, out, batch);
  (void)n_in; (void)out_size; (void)ws_size;
}